// GNN_Model_52896817217995
// MI455X (gfx1250) — compile-verified
//
#include <hip/hip_runtime.h>
#include <hip/hip_bf16.h>
#include <stdint.h>

// ---------------------------------------------------------------------------
// DevConv GNN, fused.  N=8192 vertices, dense bool adjacency (~0.4%).
// Key identity: neighbor aggregation depends only on nodes/Wtheta, so all
// three layers' neighbor-mins are computed in ONE pass over adj (64MB -> the
// whole problem is one HBM stream, ~2.9us at 23.3TB/s).
// ---------------------------------------------------------------------------

#define NVERT  8192
#define STRIDE 72          // 66 channels (1 + 64 + 1) padded to 72 (288B, 16B-aligned rows)
#define BIGF   1e30f

typedef __attribute__((ext_vector_type(2))) float v2f;
typedef __attribute__((ext_vector_type(8))) float v8f;

__device__ __forceinline__ void fmin4(float4& a, const float4 b) {
    a.x = fminf(a.x, b.x);
    a.y = fminf(a.y, b.y);
    a.z = fminf(a.z, b.z);
    a.w = fminf(a.w, b.w);
}

// ---------------------------------------------------------------------------
// Kernel A: tp_all[i][0]    = nodes[i] . Wtheta1
//           tp_all[i][1+c]  = (nodes @ Wtheta2)[i][c]   (via v_wmma_f32_16x16x4_f32)
//           tp_all[i][65]   = nodes[i] . Wtheta3
// One wave per 16-row tile; K=3 padded to 4; 4 col-tiles of 16 for the 64 cols.
// ---------------------------------------------------------------------------
__global__ __launch_bounds__(256) void tp_precompute(
    const float* __restrict__ nodes,
    const float* __restrict__ Wt1,
    const float* __restrict__ Wt2,
    const float* __restrict__ Wt3,
    float* __restrict__ tp)
{
    const int gtid = blockIdx.x * blockDim.x + threadIdx.x;
    const int wave = gtid >> 5;           // 512 waves total, 16 rows each
    const int lane = threadIdx.x & 31;
    const int l15  = lane & 15;
    const int hi   = lane >> 4;           // 0: lanes 0-15, 1: lanes 16-31
    const int r0   = wave * 16;

    // A fragment (16x4 f32): lanes 0-15 hold K=0,1 ; lanes 16-31 hold K=2,3(pad)
    const int ra = r0 + l15;
    v2f a;
    if (hi == 0) { a.x = nodes[ra * 3 + 0]; a.y = nodes[ra * 3 + 1]; }
    else         { a.x = nodes[ra * 3 + 2]; a.y = 0.0f;              }

    #pragma unroll
    for (int t = 0; t < 4; ++t) {
        const int col = t * 16 + l15;
        // B fragment (4x16 f32): VGPR0 = rows K=0 (lanes 0-15) / K=2 (16-31),
        //                        VGPR1 = rows K=1 / K=3(pad)
        v2f b;
        if (hi == 0) { b.x = Wt2[0 * 64 + col]; b.y = Wt2[1 * 64 + col]; }
        else         { b.x = Wt2[2 * 64 + col]; b.y = 0.0f;              }

        v8f c = {};
        c = __builtin_amdgcn_wmma_f32_16x16x4_f32(
                /*neg_a=*/false, a, /*neg_b=*/false, b,
                /*c_mod=*/(short)0, c, /*reuse_a=*/false, /*reuse_b=*/false);

        // D (16x16 f32): VGPR r -> row M=r (lanes 0-15, N=lane) / M=r+8 (lanes 16-31)
        #pragma unroll
        for (int r = 0; r < 8; ++r) {
            const int row = r0 + r + hi * 8;
            tp[(size_t)row * STRIDE + 1 + col] = c[r];
        }
    }

    // channels 0 (tp1) and 65 (tp3): lanes 0-15 handle rows r0..r0+15
    if (hi == 0) {
        const float n0 = nodes[ra * 3 + 0];
        const float n1 = nodes[ra * 3 + 1];
        const float n2 = nodes[ra * 3 + 2];
        tp[(size_t)ra * STRIDE + 0]  = n0 * Wt1[0] + n1 * Wt1[1] + n2 * Wt1[2];
        tp[(size_t)ra * STRIDE + 65] = n0 * Wt3[0] + n1 * Wt3[1] + n2 * Wt3[2];
    }
}

// ---------------------------------------------------------------------------
// Kernel B: one wave32 per vertex row. Stream adj row coalesced (uint4/lane),
// min-accumulate 68 channels in registers, shuffle-reduce, then finish the
// whole 3-layer per-vertex chain and write sigmoid output.
// ---------------------------------------------------------------------------
__global__ __launch_bounds__(256) void devconv_fused(
    const uint8_t* __restrict__ adj,
    const float*   __restrict__ tp,
    const float*   __restrict__ x,
    const float* __restrict__ Wphi1, const float* __restrict__ bphi1, const float* __restrict__ btheta1,
    const float* __restrict__ Wphi2, const float* __restrict__ bphi2, const float* __restrict__ btheta2,
    const float* __restrict__ Wphi3, const float* __restrict__ bphi3, const float* __restrict__ btheta3,
    float* __restrict__ out)
{
    __shared__ float sW2[64], sB2[64], sT2[64], sW3[64];
    __shared__ __align__(16) float smin[8][68];

    const int tid = threadIdx.x;
    if (tid < 64) {
        sW2[tid] = Wphi2[tid];
        sB2[tid] = bphi2[tid];
        sT2[tid] = btheta2[tid];
        sW3[tid] = Wphi3[tid];
    }
    __syncthreads();

    const int w    = tid >> 5;
    const int lane = tid & 31;
    const int row  = blockIdx.x * 8 + w;          // 1024 blocks * 8 waves = 8192 rows

    // 68-channel min accumulators (17 x float4), kept in registers
    float4 mn[17];
    #pragma unroll
    for (int u = 0; u < 17; ++u) mn[u] = make_float4(BIGF, BIGF, BIGF, BIGF);

    const uint4* rowp = (const uint4*)(adj + (size_t)row * NVERT);

    #pragma unroll 1
    for (int it = 0; it < 16; ++it) {
        if (it + 4 < 16) __builtin_prefetch(&rowp[(it + 4) * 32 + lane], 0, 0);
        const uint4 v = rowp[it * 32 + lane];
        if (v.x | v.y | v.z | v.w) {                       // ~6% of 16B chunks
            const int jb = (it * 32 + lane) * 16;
            #pragma unroll 1
            for (int q = 0; q < 4; ++q) {
                uint32_t dq = (q == 0) ? v.x : (q == 1) ? v.y : (q == 2) ? v.z : v.w;
                if (!dq) continue;
                #pragma unroll 1
                for (int k = 0; k < 4; ++k) {
                    if (dq & 0xFFu) {
                        const float4* p = (const float4*)(tp + (size_t)(jb + q * 4 + k) * STRIDE);
                        #pragma unroll
                        for (int u = 0; u < 17; ++u) { float4 t = p[u]; fmin4(mn[u], t); }
                    }
                    dq >>= 8;
                }
            }
        }
    }

    // wave32 cross-lane min (xor butterfly): all lanes end with the row mins
    #pragma unroll
    for (int off = 16; off > 0; off >>= 1) {
        #pragma unroll
        for (int u = 0; u < 17; ++u) {
            mn[u].x = fminf(mn[u].x, __shfl_xor(mn[u].x, off, 32));
            mn[u].y = fminf(mn[u].y, __shfl_xor(mn[u].y, off, 32));
            mn[u].z = fminf(mn[u].z, __shfl_xor(mn[u].z, off, 32));
            mn[u].w = fminf(mn[u].w, __shfl_xor(mn[u].w, off, 32));
        }
    }

    // park mins in LDS so lanes can index their channels without register spills
    if (lane == 0) {
        #pragma unroll
        for (int u = 0; u < 17; ++u) *(float4*)&smin[w][4 * u] = mn[u];
    }
    asm volatile("s_wait_dscnt 0" ::: "memory");

    // -------- fused per-vertex 3-layer chain --------
    const float* tpr = tp + (size_t)row * STRIDE;

    // layer 1 (C=1): h1 = relu(x*Wphi1 + bphi1 + tp1 - min1 + btheta1)
    const float agg1 = tpr[0] - smin[w][0] + btheta1[0];
    const float h1   = fmaxf(x[row] * Wphi1[0] + bphi1[0] + agg1, 0.0f);

    // layer 2 (C=64, 2 channels per lane) fused with layer-3 GEMV h2 . Wphi3
    float part = 0.0f;
    #pragma unroll
    for (int hh = 0; hh < 2; ++hh) {
        const int c    = lane + 32 * hh;
        const float a2 = tpr[1 + c] - smin[w][1 + c] + sT2[c];
        const float h2 = fmaxf(h1 * sW2[c] + sB2[c] + a2, 0.0f);
        part += h2 * sW3[c];
    }
    #pragma unroll
    for (int off = 16; off > 0; off >>= 1) part += __shfl_xor(part, off, 32);

    if (lane == 0) {
        const float agg3 = tpr[65] - smin[w][65] + btheta3[0];
        const float z    = part + bphi3[0] + agg3;
        out[row] = 1.0f / (1.0f + expf(-z));
    }
}

// ---------------------------------------------------------------------------
extern "C" void kernel_launch(void* const* d_in, const int* in_sizes, int n_in,
                              void* d_out, int out_size, void* d_ws, size_t ws_size,
                              hipStream_t stream)
{
    const float*   x       = (const float*)d_in[0];
    const float*   nodes   = (const float*)d_in[1];
    const uint8_t* adj     = (const uint8_t*)d_in[2];
    const float*   Wphi1   = (const float*)d_in[3];
    const float*   bphi1   = (const float*)d_in[4];
    const float*   Wtheta1 = (const float*)d_in[5];
    const float*   btheta1 = (const float*)d_in[6];
    const float*   Wphi2   = (const float*)d_in[7];
    const float*   bphi2   = (const float*)d_in[8];
    const float*   Wtheta2 = (const float*)d_in[9];
    const float*   btheta2 = (const float*)d_in[10];
    const float*   Wphi3   = (const float*)d_in[11];
    const float*   bphi3   = (const float*)d_in[12];
    const float*   Wtheta3 = (const float*)d_in[13];
    const float*   btheta3 = (const float*)d_in[14];

    float* tp = (float*)d_ws;   // NVERT * STRIDE * 4 bytes = 2.36 MB

    // 512 waves, one 16-row WMMA tile each
    tp_precompute<<<64, 256, 0, stream>>>(nodes, Wtheta1, Wtheta2, Wtheta3, tp);

    // one wave per vertex row: 8192 waves
    devconv_fused<<<NVERT / 8, 256, 0, stream>>>(
        adj, tp, x,
        Wphi1, bphi1, btheta1,
        Wphi2, bphi2, btheta2,
        Wphi3, bphi3, btheta3,
        (float*)d_out);
}